// Encoder_44049184588152
// MI455X (gfx1250) — compile-verified
//
#include <hip/hip_runtime.h>
#include <hip/hip_bf16.h>
#include <math.h>

typedef __attribute__((ext_vector_type(2))) float v2f;
typedef __attribute__((ext_vector_type(8))) float v8f;

#define B_   128
#define T_   128
#define N_   128
#define H_   512
#define LDSROW 642   // 640 (N + H) padded so 16 A-rows hit distinct LDS banks

__device__ __forceinline__ float sigmoidf_(float x) {
    return 1.0f / (1.0f + __expf(-x));
}

// -----------------------------------------------------------------------------
// Kernel 1: time-invariant attention (softmax shift-invariance) + input_weighted
// grid = B, block = N (=128 threads). attn[b,:] -> workspace for kernel 2.
// -----------------------------------------------------------------------------
__global__ void __launch_bounds__(128) attn_kernel(
    const float* __restrict__ x,       // (B,T,N)
    const float* __restrict__ attn_W,  // (2H+T, 1)
    const float* __restrict__ attn_b,  // (1,)
    float* __restrict__ out1,          // (B,N,T) input_weighted
    float* __restrict__ attn_ws)       // (B,N)
{
    __shared__ float WxS[T_];
    __shared__ float red[N_];
    const int b = blockIdx.x;
    const int n = threadIdx.x;

    WxS[n] = attn_W[2 * H_ + n];       // Wx slice
    __syncthreads();

    const float* xb = x + (size_t)b * T_ * N_;
    float s = attn_b[0];
    for (int t = 0; t < T_; ++t) s += xb[t * N_ + n] * WxS[t];

    // softmax over n (the h@Wh + c@Wc terms are scalar shifts -> drop out)
    red[n] = s; __syncthreads();
    for (int off = 64; off > 0; off >>= 1) {
        if (n < off) { float o = red[n + off]; if (o > red[n]) red[n] = o; }
        __syncthreads();
    }
    const float mx = red[0]; __syncthreads();
    const float e = __expf(s - mx);
    red[n] = e; __syncthreads();
    for (int off = 64; off > 0; off >>= 1) {
        if (n < off) red[n] += red[n + off];
        __syncthreads();
    }
    const float a = e / red[0];
    attn_ws[b * N_ + n] = a;

    // input_weighted[b,n,t] = attn[b,n] * x[b,t,n]  (each lane streams a T-row)
    float* o1 = out1 + (size_t)b * N_ * T_ + (size_t)n * T_;
    for (int t = 0; t < T_; ++t) o1[t] = a * xb[t * N_ + n];
}

// -----------------------------------------------------------------------------
// Kernel 2: 128-step LSTM recurrence, WMMA f32 16x16x4.
// grid = 8 (batch tiles of 16 rows), block = 512 (16 waves).
// Per step: gates(16x2048) = A(16x640) @ W(640x2048),  A = [w_in_t | h].
// Wave w owns 32 columns of each gate block (i/f/g/o) -> 8 n-tiles, so the
// cell update is elementwise on its own D fragments; c stays in registers.
// -----------------------------------------------------------------------------
__global__ void __launch_bounds__(512) lstm_kernel(
    const float* __restrict__ x,       // (B,T,N)
    const float* __restrict__ W_ih,    // (4H, N) row-major
    const float* __restrict__ W_hh,    // (4H, H) row-major
    const float* __restrict__ b_ih,    // (4H,)
    const float* __restrict__ b_hh,    // (4H,)
    const float* __restrict__ attn_ws, // (B,N)
    float* __restrict__ out2)          // (B,T,H) input_encoded
{
    __shared__ float ab[16][LDSROW];   // [m][0..127]=w_in_t, [m][128..639]=h
    __shared__ float attn_s[16][N_];

    const int tid  = threadIdx.x;
    const int w    = tid >> 5;         // wave id 0..15
    const int lane = tid & 31;
    const int b0   = blockIdx.x * 16;  // batch tile base

    // stage attention tile; zero h state
    for (int e = tid; e < 16 * N_; e += 512) {
        int m = e >> 7, n = e & 127;
        attn_s[m][n] = attn_ws[(b0 + m) * N_ + n];
    }
    for (int e = tid; e < 16 * H_; e += 512) {
        int m = e >> 9, j = e & 511;
        ab[m][128 + j] = 0.0f;
    }

    // per-lane fragment coordinates (ISA 7.12.2 layouts)
    const int mrow = lane & 15;        // A row / B,D column within tile
    const int khi  = lane >> 4;        // A/B K half
    const int kk0  = khi * 2;

    // tile tt = g*2+half owns gate-block g, columns [w*32+half*16, +16)
    int   colb[8];
    float bsum[8];
    #pragma unroll
    for (int tt = 0; tt < 8; ++tt) {
        int g = tt >> 1, half = tt & 1;
        colb[tt] = g * 512 + w * 32 + half * 16;
        int col  = colb[tt] + mrow;
        bsum[tt] = b_ih[col] + b_hh[col];   // time-invariant, hoisted
    }

    const v8f vzero = {};
    v8f cstate[2];                     // c for the wave's 32 h-columns
    cstate[0] = vzero; cstate[1] = vzero;

    for (int t = 0; t < T_; ++t) {
        // stage w_in_t[m][n] = attn[m][n] * x[b0+m, t, n]  (coalesced over n)
        for (int e = tid; e < 16 * N_; e += 512) {
            int m = e >> 7, n = e & 127;
            ab[m][n] = attn_s[m][n] * x[(size_t)(b0 + m) * T_ * N_ + (size_t)t * N_ + n];
        }
        __syncthreads();               // staging + prev h-writes visible

        v8f acc[8];
        #pragma unroll
        for (int tt = 0; tt < 8; ++tt) acc[tt] = vzero;

        // K-part 1: w_in (K=0..127) against W_ih
        for (int k = 0; k < N_; k += 4) {
            v2f a = *(const v2f*)&ab[mrow][k + kk0];
            #pragma unroll
            for (int tt = 0; tt < 8; ++tt) {
                int col = colb[tt] + mrow;
                v2f bf = *(const v2f*)&W_ih[(size_t)col * N_ + k + kk0];
                acc[tt] = __builtin_amdgcn_wmma_f32_16x16x4_f32(
                    false, a, false, bf, (short)0, acc[tt], false, false);
            }
        }
        // K-part 2: h (K=128..639) against W_hh
        for (int k = 0; k < H_; k += 4) {
            v2f a = *(const v2f*)&ab[mrow][128 + k + kk0];
            #pragma unroll
            for (int tt = 0; tt < 8; ++tt) {
                int col = colb[tt] + mrow;
                v2f bf = *(const v2f*)&W_hh[(size_t)col * H_ + k + kk0];
                acc[tt] = __builtin_amdgcn_wmma_f32_16x16x4_f32(
                    false, a, false, bf, (short)0, acc[tt], false, false);
            }
        }
        __syncthreads();               // all GEMM reads of ab done

        // cell update: elementwise on D fragments (same lane/vgpr position in
        // each gate block). D layout: vgpr v, lanes0-15 -> M=v, lanes16-31 -> M=v+8.
        #pragma unroll
        for (int half = 0; half < 2; ++half) {
            v8f I = acc[half], F = acc[2 + half], G = acc[4 + half], O = acc[6 + half];
            #pragma unroll
            for (int v = 0; v < 8; ++v) {
                float gi = I[v] + bsum[half];
                float gf = F[v] + bsum[2 + half];
                float gg = G[v] + bsum[4 + half];
                float go = O[v] + bsum[6 + half];
                float c2 = sigmoidf_(gf) * cstate[half][v] + sigmoidf_(gi) * tanhf(gg);
                float h2 = sigmoidf_(go) * tanhf(c2);
                cstate[half][v] = c2;
                int m = v + (khi << 3);
                int j = w * 32 + half * 16 + mrow;
                ab[m][128 + j] = h2;   // h for next step's A fragment
                out2[(size_t)(b0 + m) * T_ * H_ + (size_t)t * H_ + j] = h2;
            }
        }
        // no barrier needed here: next staging touches ab[*][0..127] only,
        // and the post-staging barrier orders these h-writes vs next GEMM.
    }
}

extern "C" void kernel_launch(void* const* d_in, const int* in_sizes, int n_in,
                              void* d_out, int out_size, void* d_ws, size_t ws_size,
                              hipStream_t stream) {
    (void)in_sizes; (void)n_in; (void)out_size; (void)ws_size;
    const float* x      = (const float*)d_in[0];
    const float* attn_W = (const float*)d_in[1];
    const float* attn_b = (const float*)d_in[2];
    const float* W_ih   = (const float*)d_in[3];
    const float* W_hh   = (const float*)d_in[4];
    const float* b_ih   = (const float*)d_in[5];
    const float* b_hh   = (const float*)d_in[6];

    float* out1 = (float*)d_out;                       // input_weighted (B,N,T)
    float* out2 = out1 + (size_t)B_ * N_ * T_;         // input_encoded  (B,T,H)
    float* attn_ws = (float*)d_ws;                     // (B,N) attention weights

    attn_kernel<<<B_, 128, 0, stream>>>(x, attn_W, attn_b, out1, attn_ws);
    lstm_kernel<<<8, 512, 0, stream>>>(x, W_ih, W_hh, b_ih, b_hh, attn_ws, out2);
}